// SmoothedCrossEntropyLoss_19705309954423
// MI455X (gfx1250) — compile-verified
//
#include <hip/hip_runtime.h>
#include <hip/hip_bf16.h>

// Problem constants (reference: N = V = 8192, fp32 inputs, scalar fp32 output).
#define ALPHA   0.154f
#define VCLS    8192          // classes per row
#define TPB     256           // 8 waves of 32
#define NWAVES  (TPB / 32)
#define F4PT    (VCLS / (TPB * 4))   // float4 elements per thread = 8

typedef float v2f __attribute__((ext_vector_type(2)));
typedef float v8f __attribute__((ext_vector_type(8)));

// ---------------------------------------------------------------------------
// Wave-level (32-lane) sum reduction.
// Preferred path: V_WMMA_F32_16X16X4_F32 (full fp32 matrix op, gfx1250).
//   A (16x4, 2 VGPRs/lane): lanes 0-15 hold A[m][0],A[m][1]; lanes 16-31 hold
//   A[m][2],A[m][3].  Setting both VGPRs to the lane value v gives
//   A[m][0..1] = v_m, A[m][2..3] = v_{m+16}.  With B = ones:
//       D[m][n] = 2*(v_m + v_{m+16})  for every n.
//   Lane j (<16) holds D[0..7][j] in its 8 D VGPRs, lane j+16 holds D[8..15][j],
//   so (sum of own 8 VGPRs) + (xor-16 partner) = 2 * (full 32-lane sum).
// ---------------------------------------------------------------------------
__device__ __forceinline__ float wave_sum(float v) {
#if __has_builtin(__builtin_amdgcn_wmma_f32_16x16x4_f32)
    v2f a; a[0] = v;    a[1] = v;
    v2f b; b[0] = 1.0f; b[1] = 1.0f;
    v8f c = {};
    v8f d = __builtin_amdgcn_wmma_f32_16x16x4_f32(
        /*neg_a=*/false, a, /*neg_b=*/false, b,
        /*c_mod=*/(short)0, c, /*reuse_a=*/false, /*reuse_b=*/false);
    float r = ((d[0] + d[1]) + (d[2] + d[3])) + ((d[4] + d[5]) + (d[6] + d[7]));
    r += __shfl_xor(r, 16, 32);
    return r * 0.5f;
#else
    for (int m = 16; m > 0; m >>= 1) v += __shfl_xor(v, m, 32);
    return v;
#endif
}

__device__ __forceinline__ float wave_max(float v) {
    for (int m = 16; m > 0; m >>= 1) v = fmaxf(v, __shfl_xor(v, m, 32));
    return v;
}

// ---------------------------------------------------------------------------
// One workgroup per row.  Row (32 KB) is held entirely in registers:
// single pass over HBM (256 MB total -> ~11 us at 23.3 TB/s).
// ---------------------------------------------------------------------------
__global__ void __launch_bounds__(TPB)
sce_row_kernel(const float* __restrict__ x, float* __restrict__ partial) {
    const int row  = blockIdx.x;
    const int tid  = threadIdx.x;
    const int lane = tid & 31;
    const int wid  = tid >> 5;

    __shared__ float red[NWAVES];

    // 1) Load the whole row into registers (coalesced b128 loads, 8 in flight).
    const float4* rp = (const float4*)(x + (size_t)row * (size_t)VCLS);
    float4 d[F4PT];
#pragma unroll
    for (int k = 0; k < F4PT; ++k) d[k] = rp[tid + k * TPB];

    // 2) Row max (block reduction).
    float m = -3.402823466e38f;
#pragma unroll
    for (int k = 0; k < F4PT; ++k)
        m = fmaxf(m, fmaxf(fmaxf(d[k].x, d[k].y), fmaxf(d[k].z, d[k].w)));
    m = wave_max(m);
    if (lane == 0) red[wid] = m;
    __syncthreads();
    float bm = red[0];
#pragma unroll
    for (int i = 1; i < NWAVES; ++i) bm = fmaxf(bm, red[i]);
    __syncthreads();

    // 3) e = exp(x - max) in place; sum of exps (WMMA wave reduction).
    float s = 0.0f;
#pragma unroll
    for (int k = 0; k < F4PT; ++k) {
        d[k].x = __expf(d[k].x - bm);
        d[k].y = __expf(d[k].y - bm);
        d[k].z = __expf(d[k].z - bm);
        d[k].w = __expf(d[k].w - bm);
        s += (d[k].x + d[k].y) + (d[k].z + d[k].w);
    }
    float wsum = wave_sum(s);
    if (lane == 0) red[wid] = wsum;
    __syncthreads();
    float bs = red[0];
#pragma unroll
    for (int i = 1; i < NWAVES; ++i) bs += red[i];
    __syncthreads();

    // 4) acc = sum p * log(c - p),  p = e / rowsum.
    const float inv = 1.0f / bs;
    const float c   = 1.0f - ALPHA + ALPHA / (float)VCLS;
    float acc = 0.0f;
#pragma unroll
    for (int k = 0; k < F4PT; ++k) {
        float p0 = d[k].x * inv;
        float p1 = d[k].y * inv;
        float p2 = d[k].z * inv;
        float p3 = d[k].w * inv;
        acc += p0 * __logf(c - p0);
        acc += p1 * __logf(c - p1);
        acc += p2 * __logf(c - p2);
        acc += p3 * __logf(c - p3);
    }
    float wacc = wave_sum(acc);
    if (lane == 0) red[wid] = wacc;
    __syncthreads();
    if (tid == 0) {
        float b = red[0];
#pragma unroll
        for (int i = 1; i < NWAVES; ++i) b += red[i];
        partial[row] = b;
    }
}

// ---------------------------------------------------------------------------
// Deterministic final reduction of per-row partials -> scalar output.
// ---------------------------------------------------------------------------
__global__ void __launch_bounds__(256)
sce_final_kernel(const float* __restrict__ partial, float* __restrict__ out, int n) {
    __shared__ float sm[256];
    float s = 0.0f;
    for (int i = threadIdx.x; i < n; i += 256) s += partial[i];
    sm[threadIdx.x] = s;
    __syncthreads();
    for (int off = 128; off > 0; off >>= 1) {
        if (threadIdx.x < off) sm[threadIdx.x] += sm[threadIdx.x + off];
        __syncthreads();
    }
    if (threadIdx.x == 0) out[0] = -sm[0];
}

extern "C" void kernel_launch(void* const* d_in, const int* in_sizes, int n_in,
                              void* d_out, int out_size, void* d_ws, size_t ws_size,
                              hipStream_t stream) {
    const float* inputs = (const float*)d_in[0];
    // d_in[1] (targets) is mathematically unused by the reference result.
    const int total = in_sizes[0];
    const int nrows = total / VCLS;

    float* partial = (float*)d_ws;   // nrows floats = 32 KB scratch

    sce_row_kernel<<<nrows, TPB, 0, stream>>>(inputs, partial);
    sce_final_kernel<<<1, 256, 0, stream>>>(partial, (float*)d_out, nrows);
}